// FilterDetections_76862734729357
// MI455X (gfx1250) — compile-verified
//
#include <hip/hip_runtime.h>
#include <hip/hip_bf16.h>
#include <math.h>

#define NANCH 4096
#define NCLS  8
#define KHYP  11          // POSE_HYPS + 1
#define MAXDET 100
#define CHUNK 1024        // anchors staged per LDS tile

// ---------------------------------------------------------------------------
// Kernel 1: per-class cluster-head assignment.
//   For each valid anchor c: max_col[c] = lowest valid j with IoU(c,j) > 0.5
//   (guaranteed <= c via the diagonal), key[c] = (1 - IoU) * conf[c].
//   Boxes are staged global->LDS with CDNA5 async-to-LDS copies (ASYNCcnt),
//   double-buffered so the copy of tile ch+1 overlaps the scan of tile ch.
//   The j-scan walks only set bits of the validity mask (ctz/Kernighan).
// ---------------------------------------------------------------------------
__global__ __launch_bounds__(256) void k_heads(
    const float* __restrict__ boxes,      // [N][C*4]
    const float* __restrict__ clsf,       // [N][C]
    const float* __restrict__ conf,       // [N][C]
    int*   __restrict__ maxcol,           // [C][N]
    float* __restrict__ keyv,             // [C][N]
    int*   __restrict__ mcls)             // [C]
{
  const int tid  = threadIdx.x;
  const int cblk = blockIdx.x;            // 0..15 : which 256-anchor slice
  const int k    = blockIdx.y;            // class
  const int c    = cblk * 256 + tid;      // my anchor

  __shared__ float4   sbox[2][CHUNK];     // double-buffered box tiles (32 KB)
  __shared__ unsigned vmask[NANCH / 32];
  __shared__ int      mcount;

  if (tid == 0) mcount = 0;
  __syncthreads();

  // Valid-anchor bitmask for the whole class (score > 0.5), plus m = popcount.
  if (tid < NANCH / 32) {
    unsigned w = 0;
    const int base = tid * 32;
    for (int b = 0; b < 32; ++b)
      if (clsf[(size_t)(base + b) * NCLS + k] > 0.5f) w |= (1u << b);
    vmask[tid] = w;
    atomicAdd(&mcount, __popc(w));
  }
  __syncthreads();
  if (cblk == 0 && tid == 0) mcls[k] = mcount;

  const bool myvalid = (vmask[c >> 5] >> (c & 31)) & 1u;

  const float4 mb = *(const float4*)(boxes + (size_t)c * (NCLS * 4) + k * 4);
  const float a0 = mb.x, a1 = mb.y, a2 = mb.z, a3 = mb.w;
  const float areaA = (a2 - a0 + 1.0f) * (a3 - a1 + 1.0f);

  // async stage of one 1024-anchor tile: 4 x B128 per thread
  auto issue_tile = [&](int ch) {
    const int buf = ch & 1;
#pragma unroll
    for (int q = 0; q < 4; ++q) {
      const int a = ch * CHUNK + q * 256 + tid;
      const float* gp = boxes + (size_t)a * (NCLS * 4) + k * 4;
      const unsigned laddr = (unsigned)(unsigned long long)(uintptr_t)(&sbox[buf][q * 256 + tid]);
      asm volatile("global_load_async_to_lds_b128 %0, %1, off"
                   :: "v"(laddr), "v"(gp) : "memory");
    }
  };

  int   fj  = -1;
  float fov = 0.0f;
  const int nch = cblk / 4 + 1;           // tiles needed to cover j <= c

  issue_tile(0);
  asm volatile("s_wait_asynccnt 0" ::: "memory");
  __syncthreads();

  for (int ch = 0; ch < nch; ++ch) {
    if (ch + 1 < nch) issue_tile(ch + 1); // overlap copy with scan

    if (myvalid && fj < 0) {
      const int j0   = ch * CHUNK;                 // multiple of 32
      const int jend = min(j0 + CHUNK - 1, c);     // diagonal bound
      const int buf  = ch & 1;
      const int wlo  = j0 >> 5;
      const int whi  = jend >> 5;
      for (int wi = wlo; wi <= whi && fj < 0; ++wi) {
        unsigned w = vmask[wi];
        if (wi == whi) {                           // clip bits beyond jend
          const int hb = jend & 31;
          w &= (hb == 31) ? 0xFFFFFFFFu : ((2u << hb) - 1u);
        }
        while (w) {                                // walk set bits only
          const int b = __builtin_ctz(w);
          w &= w - 1u;
          const int j = (wi << 5) + b;
          const float4 bb = sbox[buf][j - j0];
          const float x1 = fmaxf(a0, bb.x), y1 = fmaxf(a1, bb.y);
          const float x2 = fminf(a2, bb.z), y2 = fminf(a3, bb.w);
          const float ww = x2 - x1 + 1.0f, hh = y2 - y1 + 1.0f;
          if (ww > 0.0f && hh > 0.0f) {
            const float inter = ww * hh;
            const float areaB = (bb.z - bb.x + 1.0f) * (bb.w - bb.y + 1.0f);
            const float den   = areaA + areaB - inter;
            const float ov    = (den == 0.0f) ? 0.0f : inter / den;
            if (ov > 0.5f) { fj = j; fov = ov; break; }
          }
        }
      }
    }

    asm volatile("s_wait_asynccnt 0" ::: "memory");
    __syncthreads();
  }

  const int o = k * NANCH + c;
  if (myvalid) {
    maxcol[o] = fj;                                     // fj >= 0 guaranteed
    keyv[o]   = (1.0f - fov) * conf[(size_t)c * NCLS + k];
  } else {
    maxcol[o] = -1;
    keyv[o]   = 0.0f;
  }
}

// ---------------------------------------------------------------------------
// Kernel 2: per head row r, gather members {c : max_col[c]==r, key!=0},
// keep the min(cnt, min(11, m)) smallest (key, idx) and average poses/boxes.
// ---------------------------------------------------------------------------
__global__ __launch_bounds__(256) void k_cluster(
    const float* __restrict__ boxes,      // [N][C*4]
    const float* __restrict__ poses,      // [N][C*12]
    const int*   __restrict__ maxcol,
    const float* __restrict__ keyv,
    const int*   __restrict__ mcls,
    int*   __restrict__ surv,             // [C*N]
    float* __restrict__ avgP,             // [C*N][12]
    float* __restrict__ avgB)             // [C*N][4]
{
  const int tid = threadIdx.x;
  const int k   = blockIdx.y;
  const int r   = blockIdx.x * 256 + tid;

  __shared__ int   smax[NANCH];
  __shared__ float skey[NANCH];
  for (int i = tid; i < NANCH; i += 256) {
    smax[i] = maxcol[k * NANCH + i];
    skey[i] = keyv[k * NANCH + i];
  }
  __syncthreads();

  float bk[KHYP];
  int   bi[KHYP];
#pragma unroll
  for (int t = 0; t < KHYP; ++t) { bk[t] = 3.0e38f; bi[t] = 0x7fffffff; }

  int cnt = 0;
  for (int cc = 0; cc < NANCH; ++cc) {
    if (smax[cc] == r) {
      const float kv = skey[cc];
      if (kv != 0.0f) {                   // bc==0 entries become 1000 -> masked
        ++cnt;
        if (kv < bk[KHYP - 1] ||
            (kv == bk[KHYP - 1] && cc < bi[KHYP - 1])) {
          float ck = kv; int ci = cc;     // bubble-insert, index tie-break
#pragma unroll
          for (int t = 0; t < KHYP; ++t) {
            const bool sw = (ck < bk[t]) || (ck == bk[t] && ci < bi[t]);
            if (sw) {
              const float tk = bk[t]; const int ti = bi[t];
              bk[t] = ck; bi[t] = ci; ck = tk; ci = ti;
            }
          }
        }
      }
    }
  }

  const int cap     = min(KHYP, mcls[k]);
  const int contrib = min(cnt, cap);
  const int o       = k * NANCH + r;
  surv[o] = (contrib > 0) ? 1 : 0;

  if (contrib > 0) {
    float ps[12]; float bs[4];
#pragma unroll
    for (int q = 0; q < 12; ++q) ps[q] = 0.0f;
#pragma unroll
    for (int q = 0; q < 4; ++q)  bs[q] = 0.0f;
    for (int t = 0; t < contrib; ++t) {
      const int cc = bi[t];
      const float* pp = poses + (size_t)cc * (NCLS * 12) + k * 12;
#pragma unroll
      for (int q = 0; q < 12; ++q) ps[q] += pp[q];
      const float* bp = boxes + (size_t)cc * (NCLS * 4) + k * 4;
#pragma unroll
      for (int q = 0; q < 4; ++q)  bs[q] += bp[q];
    }
    const float inv = 1.0f / (float)contrib;
#pragma unroll
    for (int q = 0; q < 12; ++q) avgP[(size_t)o * 12 + q] = ps[q] * inv;
#pragma unroll
    for (int q = 0; q < 4; ++q)  avgB[(size_t)o * 4 + q]  = bs[q] * inv;
  }
}

// ---------------------------------------------------------------------------
// Kernel 3 (single block): prefix-sum compaction of survivors (flat order),
// write labels/idx/poses/boxes for ranks < 100, then 100 argmax passes over
// the compacted scores for out_scores. Defaults are -1 everywhere.
//   d_out layout: scores[0:100] labels[100:200] poses[200:1400]
//                 idx[1400:1500] boxes[1500:1900]
// ---------------------------------------------------------------------------
__global__ __launch_bounds__(1024) void k_final(
    const float* __restrict__ clsf,       // [N][C]
    const int*   __restrict__ surv,       // [C*N]
    const float* __restrict__ avgP,
    const float* __restrict__ avgB,
    float* __restrict__ scomp,            // [C*N] workspace
    float* __restrict__ out)
{
  const int tid = threadIdx.x;
  const int TOT = NCLS * NANCH;           // 32768
  const int E   = TOT / 1024;             // 32 entries per thread

  __shared__ int   psum[1024];
  __shared__ float rv[1024];
  __shared__ int   ri[1024];

  for (int i = tid; i < 1900; i += 1024) out[i] = -1.0f;

  const int base = tid * E;
  int cnt = 0;
  for (int e = 0; e < E; ++e) cnt += surv[base + e];
  psum[tid] = cnt;
  __syncthreads();

  for (int off = 1; off < 1024; off <<= 1) {   // Hillis-Steele inclusive scan
    const int v = (tid >= off) ? psum[tid - off] : 0;
    __syncthreads();
    psum[tid] += v;
    __syncthreads();
  }
  const int total = psum[1023];
  int rank = psum[tid] - cnt;                  // exclusive prefix

  for (int e = 0; e < E; ++e) {
    const int f = base + e;
    if (surv[f]) {
      const int kc = f >> 12;                  // class
      const int r  = f & (NANCH - 1);          // anchor
      scomp[rank] = clsf[(size_t)r * NCLS + kc];
      if (rank < MAXDET) {
        out[100 + rank]  = (float)kc;          // labels
        out[1400 + rank] = (float)r;           // anchor idx
#pragma unroll
        for (int q = 0; q < 12; ++q) out[200 + rank * 12 + q] = avgP[(size_t)f * 12 + q];
#pragma unroll
        for (int q = 0; q < 4; ++q)  out[1500 + rank * 4 + q] = avgB[(size_t)f * 4 + q];
      }
      ++rank;
    }
  }
  __threadfence_block();
  __syncthreads();

  // top-100 survivor scores: value desc, tie -> lower (flat) index
  const int KK = (total < MAXDET) ? total : MAXDET;
  for (int kk = 0; kk < KK; ++kk) {
    float bv = -INFINITY; int bidx = 0x7fffffff;
    for (int i = tid; i < total; i += 1024) {
      const float v = scomp[i];
      if (v > bv || (v == bv && i < bidx)) { bv = v; bidx = i; }
    }
    rv[tid] = bv; ri[tid] = bidx;
    __syncthreads();
    for (int off = 512; off > 0; off >>= 1) {
      if (tid < off) {
        const float ov = rv[tid + off]; const int oi = ri[tid + off];
        if (ov > rv[tid] || (ov == rv[tid] && oi < ri[tid])) { rv[tid] = ov; ri[tid] = oi; }
      }
      __syncthreads();
    }
    if (tid == 0) { out[kk] = rv[0]; scomp[ri[0]] = -INFINITY; }
    __threadfence_block();
    __syncthreads();
  }
}

// ---------------------------------------------------------------------------
extern "C" void kernel_launch(void* const* d_in, const int* in_sizes, int n_in,
                              void* d_out, int out_size, void* d_ws, size_t ws_size,
                              hipStream_t stream) {
  // Inputs (setup_inputs order): boxes3D (unused), boxes, classification,
  // poses, confidence — all float32.
  const float* boxes = (const float*)d_in[1];
  const float* clsf  = (const float*)d_in[2];
  const float* poses = (const float*)d_in[3];
  const float* conf  = (const float*)d_in[4];
  float* out = (float*)d_out;

  // Workspace layout (4-byte elements): ~2.6 MB total.
  int*   maxcol = (int*)d_ws;                        // 32768
  float* keyv   = (float*)d_ws + 32768;              // 32768
  int*   mcls   = (int*)d_ws + 65536;                // 8
  int*   surv   = (int*)d_ws + 65544;                // 32768
  float* avgP   = (float*)d_ws + 98312;              // 32768*12
  float* avgB   = (float*)d_ws + 491528;             // 32768*4
  float* scomp  = (float*)d_ws + 622600;             // 32768

  k_heads<<<dim3(16, NCLS), 256, 0, stream>>>(boxes, clsf, conf, maxcol, keyv, mcls);
  k_cluster<<<dim3(16, NCLS), 256, 0, stream>>>(boxes, poses, maxcol, keyv, mcls,
                                                surv, avgP, avgB);
  k_final<<<1, 1024, 0, stream>>>(clsf, surv, avgP, avgB, scomp, out);
}